// ContrastiveHead_20375324852923
// MI455X (gfx1250) — compile-verified
//
#include <hip/hip_runtime.h>
#include <cstdint>
#include <cstddef>

#define N_COLS 8192
#define TPB 256
#define VEC_PER_THREAD 8   // (8192 / 4 floats-per-vec) / 256 threads
#define NWAVES (TPB / 32)

// Issue one 16B-per-lane async global->LDS copy. The 24-bit immediate offset
// is added to BOTH the global address and the LDS address (CDNA5 ISA
// §4.4), so one base address pair serves all 8 chunks of the row.
#define ASYNC_LD16(LDSADDR, GADDR, OFF)                                        \
  asm volatile("global_load_async_to_lds_b128 %0, %1, off offset:" #OFF        \
               :: "v"(LDSADDR), "v"(GADDR) : "memory")

__global__ __launch_bounds__(TPB) void contrastive_row_kernel(
    const float* __restrict__ sim, const int* __restrict__ sel,
    float* __restrict__ losses)
{
    __shared__ float4 simLds[N_COLS / 4];     // 32 KB: sim row, then masked logits
    __shared__ float  wps[NWAVES];            // per-wave positive sums
    __shared__ int    wcnt[NWAVES];           // per-wave positive counts
    __shared__ float  wmax[NWAVES];           // per-wave neg-logit max
    __shared__ float  wsum[NWAVES];           // per-wave exp sums
    __shared__ float  sM, sPL;

    const int tid = threadIdx.x;
    const int row = blockIdx.x;
    const float NEG_INF = -__builtin_inff();
    const float INVT = 10.0f;                  // 1 / 0.1

    const float4* gsim = ((const float4*)(sim + (size_t)row * N_COLS)) + tid;
    const int4*   gsel = ((const int4*)  (sel + (size_t)row * N_COLS)) + tid;

    // Workgroup-relative LDS byte address for this lane's 16B slot.
    uint32_t ldsSim = (uint32_t)(uintptr_t)(&simLds[0]) + (uint32_t)tid * 16u;
    const void* gsimv = (const void*)gsim;

    // Kick off the 32 KB sim-row fill on the async path (ASYNCcnt-tracked;
    // 256 lanes x 16B = 4 KB per issue, 8 issues)...
    ASYNC_LD16(ldsSim, gsimv, 0);
    ASYNC_LD16(ldsSim, gsimv, 4096);
    ASYNC_LD16(ldsSim, gsimv, 8192);
    ASYNC_LD16(ldsSim, gsimv, 12288);
    ASYNC_LD16(ldsSim, gsimv, 16384);
    ASYNC_LD16(ldsSim, gsimv, 20480);
    ASYNC_LD16(ldsSim, gsimv, 24576);
    ASYNC_LD16(ldsSim, gsimv, 28672);

    // ...and concurrently stream the select row into registers (LOADcnt is a
    // separate counter, so both HBM streams are in flight at once).
    int4 c[VEC_PER_THREAD];
#pragma unroll
    for (int it = 0; it < VEC_PER_THREAD; ++it) c[it] = gsel[it * TPB];

    asm volatile("s_wait_asynccnt 0" ::: "memory");
    __syncthreads();

    // ---- Pass A: positive sum/count, negative max; write masked logits
    // back in place (-inf for positives => branch-free pass B).
    float psum = 0.0f;
    int   cnt  = 0;
    float m    = NEG_INF;
#pragma unroll
    for (int it = 0; it < VEC_PER_THREAD; ++it) {
        const int idx = tid + it * TPB;
        float4 v = simLds[idx];
        float l0, l1, l2, l3;
        if (c[it].x == 1) { psum += v.x; ++cnt; l0 = NEG_INF; } else { l0 = v.x * INVT; m = fmaxf(m, l0); }
        if (c[it].y == 1) { psum += v.y; ++cnt; l1 = NEG_INF; } else { l1 = v.y * INVT; m = fmaxf(m, l1); }
        if (c[it].z == 1) { psum += v.z; ++cnt; l2 = NEG_INF; } else { l2 = v.z * INVT; m = fmaxf(m, l2); }
        if (c[it].w == 1) { psum += v.w; ++cnt; l3 = NEG_INF; } else { l3 = v.w * INVT; m = fmaxf(m, l3); }
        simLds[idx] = make_float4(l0, l1, l2, l3);   // own slot only: race-free
    }

    // wave32 reduction, then cross-wave via LDS
#pragma unroll
    for (int o = 16; o > 0; o >>= 1) {
        psum += __shfl_down(psum, o, 32);
        cnt  += __shfl_down(cnt,  o, 32);
        m     = fmaxf(m, __shfl_down(m, o, 32));
    }
    const int wv = tid >> 5;
    if ((tid & 31) == 0) { wps[wv] = psum; wcnt[wv] = cnt; wmax[wv] = m; }
    __syncthreads();

    if (tid == 0) {
        float ps = 0.0f; int cc = 0; float mm = NEG_INF;
#pragma unroll
        for (int i = 0; i < NWAVES; ++i) { ps += wps[i]; cc += wcnt[i]; mm = fmaxf(mm, wmax[i]); }
        const float pc = (float)(cc > 0 ? cc : 1);
        const float pl = (ps / pc) * INVT;            // pos_logit
        sPL = pl;
        sM  = fmaxf(mm, pl);                          // shift M is always finite
    }
    __syncthreads();

    // ---- Pass B: sum of exp(logit - M) over the masked row from LDS.
    const float M = sM;
    float es = 0.0f;
#pragma unroll
    for (int it = 0; it < VEC_PER_THREAD; ++it) {
        float4 v = simLds[tid + it * TPB];
        es += __expf(v.x - M) + __expf(v.y - M) + __expf(v.z - M) + __expf(v.w - M);
    }
#pragma unroll
    for (int o = 16; o > 0; o >>= 1) es += __shfl_down(es, o, 32);
    if ((tid & 31) == 0) wsum[wv] = es;
    __syncthreads();

    if (tid == 0) {
        float tot = 0.0f;
#pragma unroll
        for (int i = 0; i < NWAVES; ++i) tot += wsum[i];
        const float pl   = sPL;
        const float loss = M + __logf(tot + __expf(pl - M)) - pl;  // lse - pos_logit
        losses[row] = loss;
    }
}

__global__ __launch_bounds__(256) void mean_reduce_kernel(
    const float* __restrict__ in, float* __restrict__ out, int n)
{
    __shared__ float red[256];
    float s = 0.0f;
    for (int i = threadIdx.x; i < n; i += 256) s += in[i];
    red[threadIdx.x] = s;
    __syncthreads();
    for (int o = 128; o > 0; o >>= 1) {
        if (threadIdx.x < o) red[threadIdx.x] += red[threadIdx.x + o];
        __syncthreads();
    }
    if (threadIdx.x == 0) out[0] = red[0] / (float)n;
}

extern "C" void kernel_launch(void* const* d_in, const int* in_sizes, int n_in,
                              void* d_out, int out_size, void* d_ws, size_t ws_size,
                              hipStream_t stream) {
    (void)n_in; (void)out_size; (void)ws_size;
    const float* sim = (const float*)d_in[0];
    const int*   sel = (const int*)d_in[1];
    float* losses = (float*)d_ws;                 // B floats of scratch
    const int B = in_sizes[0] / N_COLS;           // 4096

    contrastive_row_kernel<<<B, TPB, 0, stream>>>(sim, sel, losses);
    mean_reduce_kernel<<<1, 256, 0, stream>>>(losses, (float*)d_out, B);
}